// ODE_61984968016253
// MI455X (gfx1250) — compile-verified
//
#include <hip/hip_runtime.h>

// Per-sample-weight MLP: 32 -> 512 -> 512 -> 512 -> 32, SiLU x3, tanh last.
// Memory-bound streamer (~1.14 GB weights, each byte used once) built around
// V_WMMA_F32_16X16X4_F32 (full f32 precision; compute is free at 0.5 flop/B).

#define LATENT 32
#define HIDDEN 512
#define PCOUNT 558624   // per-sample flat param count

// flat-param offsets (floats): per layer W[out,in] row-major, then bias
#define W0OFF 0
#define B0OFF 16384
#define W1OFF 16896
#define B1OFF 279040
#define W2OFF 279552
#define B2OFF 541696
#define W3OFF 542208
#define B3OFF 558592

typedef float v2f __attribute__((ext_vector_type(2)));
typedef float v4f __attribute__((ext_vector_type(4)));
typedef float v8f __attribute__((ext_vector_type(8)));

__device__ __forceinline__ float act_silu(float x) {
    return x * (1.0f / (1.0f + __expf(-x)));
}

// One layer: out[r] = act( sum_i W[r,i]*h[i] + b[r] ).
// Mapping onto V_WMMA_F32_16X16X4_F32 (D = A(16x4) * B(4x16) + C):
//   A[m,k] = W[out0+m, k0+k]   (lane L: m = L&15; lanes<16 carry K0,K1; lanes>=16 K2,K3)
//   B[k,n] = h[k0+k]           (replicated over n -> every column of D is the same
//                               partial dot product; C accumulates over k0 chunks)
// Final D: lane 0 holds rows out0+0..7 in c[0..7]; lane 16 holds rows out0+8..15.
template<int DIN, int DOUT, int WOFF, int BOFF, bool LAST>
__device__ __forceinline__ void layer_matvec(const float* __restrict__ Wp,
                                             const float* __restrict__ hin,
                                             float* __restrict__ hout,
                                             float* __restrict__ gout,
                                             int lane, int wave)
{
    constexpr int TILES = DOUT / 16;
    const int m    = lane & 15;
    const int ksel = (lane >> 4) << 1;       // 0 for lanes 0-15, 2 for lanes 16-31

    for (int tile = wave; tile < TILES; tile += 16) {
        const int out0 = tile * 16;
        const float* __restrict__ wrow = Wp + WOFF + (size_t)(out0 + m) * DIN;

        v8f c = {0.f, 0.f, 0.f, 0.f, 0.f, 0.f, 0.f, 0.f};

        #pragma unroll 8
        for (int k0 = 0; k0 < DIN; k0 += 4) {
            // lane L and L+16 fetch the same 16B of this W row (coalescer dedups);
            // streamed once -> non-temporal to keep L2 for useful data.
            v4f w = __builtin_nontemporal_load((const v4f*)(wrow + k0));
            v2f a;
            a.x = (ksel == 0) ? w.x : w.z;
            a.y = (ksel == 0) ? w.y : w.w;
            v2f b = *(const v2f*)(hin + k0 + ksel);   // ds_load_b64, broadcast
            c = __builtin_amdgcn_wmma_f32_16x16x4_f32(
                    /*neg_a=*/false, a, /*neg_b=*/false, b,
                    /*c_mod=*/(short)0, c, /*reuse_a=*/false, /*reuse_b=*/false);
        }

        if (m == 0) {   // lanes 0 and 16 own the 16 rows of this tile
            const int base = out0 + ((lane >> 4) << 3);
            #pragma unroll
            for (int g = 0; g < 8; ++g) {
                float v = c[g] + Wp[BOFF + base + g];
                if (LAST) {
                    gout[base + g] = tanhf(v);
                } else {
                    hout[base + g] = act_silu(v);
                }
            }
        }
    }
}

__global__ __launch_bounds__(512) void ODE_61984968016253_kernel(
        const float* __restrict__ z,
        const float* __restrict__ params,
        float* __restrict__ out)
{
    __shared__ float hbuf0[HIDDEN];
    __shared__ float hbuf1[HIDDEN];

    const int s    = blockIdx.x;
    const int t    = threadIdx.x;
    const int lane = t & 31;
    const int wave = t >> 5;
    const float* __restrict__ Wp = params + (size_t)s * PCOUNT;

    if (t < LATENT) hbuf0[t] = z[s * LATENT + t];
    __syncthreads();

    layer_matvec<LATENT, HIDDEN, W0OFF, B0OFF, false>(Wp, hbuf0, hbuf1, nullptr, lane, wave);
    __syncthreads();
    layer_matvec<HIDDEN, HIDDEN, W1OFF, B1OFF, false>(Wp, hbuf1, hbuf0, nullptr, lane, wave);
    __syncthreads();
    layer_matvec<HIDDEN, HIDDEN, W2OFF, B2OFF, false>(Wp, hbuf0, hbuf1, nullptr, lane, wave);
    __syncthreads();
    layer_matvec<HIDDEN, LATENT, W3OFF, B3OFF, true>(Wp, hbuf1, nullptr, out + s * LATENT, lane, wave);
}

extern "C" void kernel_launch(void* const* d_in, const int* in_sizes, int n_in,
                              void* d_out, int out_size, void* d_ws, size_t ws_size,
                              hipStream_t stream) {
    // inputs: d_in[0] = t (scalar, unused), d_in[1] = z [B,32] f32,
    //         d_in[2] = params [B, PCOUNT] f32; output: [B,32] f32
    const float* z      = (const float*)d_in[1];
    const float* params = (const float*)d_in[2];
    float* out          = (float*)d_out;
    const int batch     = in_sizes[1] / LATENT;   // 512
    ODE_61984968016253_kernel<<<batch, 512, 0, stream>>>(z, params, out);
}